// BranchoBrain_16166256902666
// MI455X (gfx1250) — compile-verified
//
#include <hip/hip_runtime.h>
#include <hip/hip_bf16.h>

typedef __attribute__((ext_vector_type(2))) float v2f;
typedef __attribute__((ext_vector_type(8))) float v8f;

#define IN_CH 16
#define OUT_CH 8

// ---------------------------------------------------------------------------
// 1) deg[i] = 1.0f  (self-loop contribution)
// ---------------------------------------------------------------------------
__global__ void gcn_deg_init(float* __restrict__ deg, int n) {
    int i = blockIdx.x * blockDim.x + threadIdx.x;
    if (i < n) deg[i] = 1.0f;
}

// ---------------------------------------------------------------------------
// 2) deg[dst[e]] += 1  (float atomics: exact for counts << 2^24; L2-resident)
// ---------------------------------------------------------------------------
__global__ void gcn_deg_count(const int* __restrict__ dst,
                              float* __restrict__ deg, int e_cnt) {
    int e = blockIdx.x * blockDim.x + threadIdx.x;
    if (e < e_cnt) atomicAdd(&deg[dst[e]], 1.0f);
}

// ---------------------------------------------------------------------------
// 3) dinv[i] = rsqrt(deg[i])   (deg >= 1 always due to self loops)
// ---------------------------------------------------------------------------
__global__ void gcn_dinv(float* __restrict__ deg, int n) {
    int i = blockIdx.x * blockDim.x + threadIdx.x;
    if (i < n) {
        float d = deg[i];
        deg[i] = d > 0.0f ? __frsqrt_rn(d) : 0.0f;
    }
}

// ---------------------------------------------------------------------------
// 4) h = x @ W^T via V_WMMA_F32_16X16X4_F32 (one 16-node tile per wave),
//    fused: out[i][n] = b[n] + dinv[i]^2 * h[i][n]   (self-loop msg + bias)
//
//    A (16x4 f32): lanes 0-15 -> M=lane, VGPR0/1 = K0/K1;
//                  lanes 16-31 -> M=lane-16, VGPR0/1 = K2/K3.
//    B (4x16 f32): mirrored (lane -> N, VGPRs -> K, half-wave K split).
//    D (16x16 f32, 8 VGPRs): VGPR r, lanes 0-15 -> M=r; lanes 16-31 -> M=r+8.
//
//    Loads are unconditional (clamped indices); invalid B columns zeroed with
//    a VALU multiply; full tiles take a wave-uniform branch-free epilogue.
// ---------------------------------------------------------------------------
__global__ void gcn_gemm_wmma(const float* __restrict__ x,
                              const float* __restrict__ W,    // [8][16] row-major
                              const float* __restrict__ bias, // [8]
                              const float* __restrict__ dinv,
                              float* __restrict__ h,          // [N][8]
                              float* __restrict__ out,        // [N][8]
                              int n_nodes, int n_tiles) {
    const int lane  = threadIdx.x & 31;
    const int wave  = threadIdx.x >> 5;
    const int tile  = blockIdx.x * (blockDim.x >> 5) + wave;
    if (tile >= n_tiles) return;          // wave-uniform: EXEC stays all-1s

    const int base  = tile * 16;
    const int m     = lane & 15;          // A: row within tile; B/D: column N
    const int khalf = (lane >> 4) * 2;    // 0 (lanes 0-15) or 2 (lanes 16-31)

    // clamp A-row reads for a (theoretical) partial last tile
    int arow = base + m;
    if (arow >= n_nodes) arow = n_nodes - 1;
    const float* __restrict__ xrow = x + (size_t)arow * IN_CH;

    // A fragments: 4 x float2 (8B-aligned -> global_load_b64 o0/16/32/48)
    v2f a0 = *(const v2f*)(xrow + 0  + khalf);
    v2f a1 = *(const v2f*)(xrow + 4  + khalf);
    v2f a2 = *(const v2f*)(xrow + 8  + khalf);
    v2f a3 = *(const v2f*)(xrow + 12 + khalf);

    // B fragments: unconditional loads from clamped row, VALU-zero cols >= 8
    const int   wrow = (m < OUT_CH) ? m : 0;
    const float sel  = (m < OUT_CH) ? 1.0f : 0.0f;
    const float* __restrict__ wr = W + wrow * IN_CH + khalf;
    v2f b0 = *(const v2f*)(wr + 0);
    v2f b1 = *(const v2f*)(wr + 4);
    v2f b2 = *(const v2f*)(wr + 8);
    v2f b3 = *(const v2f*)(wr + 12);
    b0 *= sel; b1 *= sel; b2 *= sel; b3 *= sel;

    // (neg_a, A, neg_b, B, c_mod, C, reuse_a, reuse_b)
    v8f acc = {};
    acc = __builtin_amdgcn_wmma_f32_16x16x4_f32(false, a0, false, b0, (short)0, acc, false, false);
    acc = __builtin_amdgcn_wmma_f32_16x16x4_f32(false, a1, false, b1, (short)0, acc, false, false);
    acc = __builtin_amdgcn_wmma_f32_16x16x4_f32(false, a2, false, b2, (short)0, acc, false, false);
    acc = __builtin_amdgcn_wmma_f32_16x16x4_f32(false, a3, false, b3, (short)0, acc, false, false);

    // Write-out: lanes with N<8 carry valid output columns.
    const int  n     = m;
    const int  mbase = base + ((lane >> 4) * 8);
    const bool full  = (base + 16) <= n_nodes;   // wave-uniform

    if (full) {
        if (n < OUT_CH) {
            const float bn = bias[n];
            const float4 d0 = *(const float4*)(dinv + mbase);
            const float4 d1 = *(const float4*)(dinv + mbase + 4);
            const float dv[8] = {d0.x, d0.y, d0.z, d0.w, d1.x, d1.y, d1.z, d1.w};
            float* __restrict__ hp = h   + (size_t)mbase * OUT_CH + n;
            float* __restrict__ op = out + (size_t)mbase * OUT_CH + n;
#pragma unroll
            for (int r = 0; r < 8; ++r) {
                const float hv = acc[r];
                hp[r * OUT_CH] = hv;
                op[r * OUT_CH] = bn + dv[r] * dv[r] * hv;
            }
        }
    } else {
        if (n < OUT_CH) {
            const float bn = bias[n];
#pragma unroll
            for (int r = 0; r < 8; ++r) {
                const int node = mbase + r;
                if (node < n_nodes) {
                    const float hv = acc[r];
                    h[(size_t)node * OUT_CH + n] = hv;
                    const float di = dinv[node];
                    out[(size_t)node * OUT_CH + n] = bn + di * di * hv;
                }
            }
        }
    }
}

// ---------------------------------------------------------------------------
// 5) per-edge: out[dst] += dinv[src]*dinv[dst] * h[src]
//    h row gathered as two float4 (32 B); 8 global f32 atomics (L2-resident).
// ---------------------------------------------------------------------------
__global__ void gcn_scatter(const int* __restrict__ src,
                            const int* __restrict__ dst,
                            const float* __restrict__ dinv,
                            const float* __restrict__ h,
                            float* __restrict__ out, int e_cnt) {
    int e = blockIdx.x * blockDim.x + threadIdx.x;
    if (e >= e_cnt) return;
    const int s = src[e];
    const int d = dst[e];
    const float w = dinv[s] * dinv[d];
    const float4* __restrict__ hs = (const float4*)(h + (size_t)s * OUT_CH);
    const float4 h0 = hs[0];
    const float4 h1 = hs[1];
    float* __restrict__ o = out + (size_t)d * OUT_CH;
    atomicAdd(o + 0, w * h0.x);
    atomicAdd(o + 1, w * h0.y);
    atomicAdd(o + 2, w * h0.z);
    atomicAdd(o + 3, w * h0.w);
    atomicAdd(o + 4, w * h1.x);
    atomicAdd(o + 5, w * h1.y);
    atomicAdd(o + 6, w * h1.z);
    atomicAdd(o + 7, w * h1.w);
}

extern "C" void kernel_launch(void* const* d_in, const int* in_sizes, int n_in,
                              void* d_out, int out_size, void* d_ws, size_t ws_size,
                              hipStream_t stream) {
    const float* x    = (const float*)d_in[0];   // [N,16] f32
    const int*   ei   = (const int*)d_in[1];     // [2,E] int
    const float* W    = (const float*)d_in[2];   // [8,16] f32
    const float* bias = (const float*)d_in[3];   // [8] f32

    const int N = in_sizes[0] / IN_CH;
    const int E = in_sizes[1] / 2;
    const int* src = ei;
    const int* dst = ei + (size_t)E;

    // workspace: dinv [N floats] | h [N*8 floats], 256B-aligned
    float* dinv = (float*)d_ws;
    size_t h_off = (((size_t)N * sizeof(float)) + 255u) & ~(size_t)255u;
    float* h = (float*)((char*)d_ws + h_off);
    float* out = (float*)d_out;

    const int TPB = 256;

    gcn_deg_init<<<(N + TPB - 1) / TPB, TPB, 0, stream>>>(dinv, N);
    gcn_deg_count<<<(E + TPB - 1) / TPB, TPB, 0, stream>>>(dst, dinv, E);
    gcn_dinv<<<(N + TPB - 1) / TPB, TPB, 0, stream>>>(dinv, N);

    const int n_tiles = (N + 15) / 16;
    const int waves_per_block = TPB / 32;
    const int gemm_blocks = (n_tiles + waves_per_block - 1) / waves_per_block;
    gcn_gemm_wmma<<<gemm_blocks, TPB, 0, stream>>>(x, W, bias, dinv, h, out, N, n_tiles);

    gcn_scatter<<<(E + TPB - 1) / TPB, TPB, 0, stream>>>(src, dst, dinv, h, out, E);
}